// DCI_10273561772530
// MI455X (gfx1250) — compile-verified
//
#include <hip/hip_runtime.h>
#include <hip/hip_bf16.h>

// ---------------------------------------------------------------------------
// DGI / GIN forward on MI455X (gfx1250, wave32).
//   - edge aggregation: native fp32 global atomics (L2-resident bottleneck)
//   - GIN linear layers: V_WMMA_F32_16X16X4_F32, one wave per 16x32 tile
//   - scoring: cluster means over a permutation partition == flat mean over N
// ---------------------------------------------------------------------------

typedef __attribute__((ext_vector_type(2))) float v2f;
typedef __attribute__((ext_vector_type(8))) float v8f;

#define HDIM 32

__device__ __forceinline__ float softplus_f(float x) {
    // stable: max(x,0) + log1p(exp(-|x|))
    return fmaxf(x, 0.f) + log1pf(expf(-fabsf(x)));
}

// Native GLOBAL_ATOMIC_ADD_F32 (no return): relaxed, device scope.
__device__ __forceinline__ void atomic_add_f32(float* p, float v) {
    __hip_atomic_fetch_add(p, v, __ATOMIC_RELAXED, __HIP_MEMORY_SCOPE_AGENT);
}

// ---- degree count: deg[dst] += 1 ------------------------------------------
__global__ void deg_kernel(const int* __restrict__ dst, float* __restrict__ deg, int E) {
    int t = blockIdx.x * blockDim.x + threadIdx.x;
    if (t < E) atomic_add_f32(&deg[dst[t]], 1.0f);
}

// ---- deg -> deg_inv in place ----------------------------------------------
__global__ void inv_kernel(float* __restrict__ deg, int N) {
    int t = blockIdx.x * blockDim.x + threadIdx.x;
    if (t < N) {
        float d = deg[t];
        deg[t] = (d > 0.f) ? (1.0f / d) : 0.0f;
    }
}

// ---- featp[i] = feature[perm[i]] (corruption) ------------------------------
__global__ void gather_perm_kernel(const float* __restrict__ feat,
                                   const int* __restrict__ perm,
                                   float* __restrict__ outp, int N) {
    int t = blockIdx.x * blockDim.x + threadIdx.x;   // N*8 threads, float4 each
    int i = t >> 3;
    int c = (t & 7) * 4;
    if (i < N) {
        int p = perm[i];
        *(float4*)(outp + (size_t)i * HDIM + c) =
            *(const float4*)(feat + (size_t)p * HDIM + c);
    }
}

// ---- scatter-add: agg[dst] += h[src], float4 granularity -------------------
__global__ void edge_agg_kernel(const float* __restrict__ h,
                                const int* __restrict__ src,
                                const int* __restrict__ dst,
                                float* __restrict__ agg, int E) {
    int t = blockIdx.x * blockDim.x + threadIdx.x;   // E*8 threads
    int e = t >> 3;
    int c = (t & 7) * 4;
    if (e >= E) return;
    int s = src[e];
    int d = dst[e];
    float4 v = *(const float4*)(h + (size_t)s * HDIM + c);
    float* base = agg + (size_t)d * HDIM + c;
    atomic_add_f32(base + 0, v.x);
    atomic_add_f32(base + 1, v.y);
    atomic_add_f32(base + 2, v.z);
    atomic_add_f32(base + 3, v.w);
}

// ---- GIN layer: out = relu((h + agg*deg_inv) @ W + b), via fp32 WMMA -------
// One wave per 16-row tile; two 16x16 C accumulators cover the 32 output cols.
// A-fragment layout (ISA 7.12.2, 32-bit A 16x4): lanes 0-15 row M=lane, K=k0,k0+1;
// lanes 16-31 same rows, K=k0+2,k0+3. B mirrors A's K split across lane halves.
__global__ __launch_bounds__(128) void gin_layer_wmma(
        const float* __restrict__ h, const float* __restrict__ agg,
        const float* __restrict__ dinv, const float* __restrict__ W,
        const float* __restrict__ bias, float* __restrict__ out) {
    int wave = blockIdx.x * (blockDim.x >> 5) + (threadIdx.x >> 5);
    int lane = threadIdx.x & 31;
    int m0   = wave * 16;
    int nlo  = lane & 15;        // N column (within 16) / M row for A loads
    int khalf = lane >> 4;       // 0: K={k0,k0+1}, 1: K={k0+2,k0+3}
    int row  = m0 + nlo;

    float di = dinv[row];

    // Preload B fragments for all 8 K-tiles, both column halves of W (32x32).
    v2f bL[8], bR[8];
#pragma unroll
    for (int j = 0; j < 8; ++j) {
        int k = 4 * j + 2 * khalf;
        bL[j].x = W[(k)     * HDIM + nlo];
        bL[j].y = W[(k + 1) * HDIM + nlo];
        bR[j].x = W[(k)     * HDIM + 16 + nlo];
        bR[j].y = W[(k + 1) * HDIM + 16 + nlo];
    }

    v8f cL = {};
    v8f cR = {};
#pragma unroll
    for (int j = 0; j < 8; ++j) {
        int k = 4 * j + 2 * khalf;
        const float* hp = h   + (size_t)row * HDIM + k;
        const float* ap = agg + (size_t)row * HDIM + k;
        v2f a;
        a.x = hp[0] + ap[0] * di;   // fused GIN (1+agg) input
        a.y = hp[1] + ap[1] * di;
        cL = __builtin_amdgcn_wmma_f32_16x16x4_f32(false, a, false, bL[j],
                                                   (short)0, cL, false, false);
        cR = __builtin_amdgcn_wmma_f32_16x16x4_f32(false, a, false, bR[j],
                                                   (short)0, cR, false, false);
    }

    // C/D layout: VGPR r -> M = m0 + r + 8*khalf, N = nlo (+16 for right tile)
    float bvL = bias[nlo];
    float bvR = bias[16 + nlo];
#pragma unroll
    for (int r = 0; r < 8; ++r) {
        int m = m0 + r + 8 * khalf;
        float vL = cL[r] + bvL; vL = vL > 0.f ? vL : 0.f;
        float vR = cR[r] + bvR; vR = vR > 0.f ? vR : 0.f;
        out[(size_t)m * HDIM + nlo]      = vL;
        out[(size_t)m * HDIM + 16 + nlo] = vR;
    }
}

// ---- column sums of pos (for summary) --------------------------------------
__global__ void colsum_kernel(const float* __restrict__ pos,
                              float* __restrict__ colsum, int N) {
    __shared__ float sdata[256];
    int f = threadIdx.x & 31;
    int g = threadIdx.x >> 5;                 // 8 row-groups per block
    float acc = 0.f;
    int nchunks = N >> 3;
    for (int ch = blockIdx.x; ch < nchunks; ch += gridDim.x) {
        int r = ch * 8 + g;
        acc += pos[(size_t)r * HDIM + f];
    }
    sdata[threadIdx.x] = acc;
    __syncthreads();
    for (int off = 128; off >= 32; off >>= 1) {
        if (threadIdx.x < off) sdata[threadIdx.x] += sdata[threadIdx.x + off];
        __syncthreads();
    }
    if (threadIdx.x < 32) atomic_add_f32(&colsum[threadIdx.x], sdata[threadIdx.x]);
}

// ---- summary = sigmoid(colsum/N); s = Wd @ summary  (single wave) ----------
__global__ void summary_s_kernel(const float* __restrict__ colsum,
                                 const float* __restrict__ Wd,
                                 float* __restrict__ svec, int N) {
    __shared__ float ssum[32];
    int i = threadIdx.x;
    float m = colsum[i] * (1.0f / (float)N);
    ssum[i] = 1.0f / (1.0f + expf(-m));
    __syncthreads();
    float acc = 0.f;
#pragma unroll
    for (int j = 0; j < 32; ++j) acc += Wd[i * HDIM + j] * ssum[j];
    svec[i] = acc;
}

// ---- loss = (1/N) * sum_i softplus(-pos_i.s) + softplus(neg_i.s) -----------
// (cluster_idx partitions all N nodes, so per-cluster means average to this)
__global__ void loss_kernel(const float* __restrict__ pos,
                            const float* __restrict__ neg,
                            const float* __restrict__ svec,
                            float* __restrict__ out, int N) {
    __shared__ float sS[32];
    __shared__ float sdata[256];
    if (threadIdx.x < 32) sS[threadIdx.x] = svec[threadIdx.x];
    __syncthreads();
    float acc = 0.f;
    for (int i = blockIdx.x * blockDim.x + threadIdx.x; i < N;
         i += gridDim.x * blockDim.x) {
        float dp = 0.f, dn = 0.f;
#pragma unroll
        for (int f = 0; f < 32; ++f) {
            dp += pos[(size_t)i * HDIM + f] * sS[f];
            dn += neg[(size_t)i * HDIM + f] * sS[f];
        }
        acc += softplus_f(-dp) + softplus_f(dn);
    }
    sdata[threadIdx.x] = acc;
    __syncthreads();
    for (int off = 128; off > 0; off >>= 1) {
        if (threadIdx.x < off) sdata[threadIdx.x] += sdata[threadIdx.x + off];
        __syncthreads();
    }
    if (threadIdx.x == 0) atomic_add_f32(out, sdata[0] * (1.0f / (float)N));
}

// ---------------------------------------------------------------------------
extern "C" void kernel_launch(void* const* d_in, const int* in_sizes, int n_in,
                              void* d_out, int out_size, void* d_ws, size_t ws_size,
                              hipStream_t stream) {
    const float* feature = (const float*)d_in[0];
    const float* W1      = (const float*)d_in[1];
    const float* b1      = (const float*)d_in[2];
    const float* W2      = (const float*)d_in[3];
    const float* b2      = (const float*)d_in[4];
    const float* Wd      = (const float*)d_in[5];
    const int*   src     = (const int*)d_in[6];
    const int*   dst     = (const int*)d_in[7];
    const int*   perm    = (const int*)d_in[8];
    // d_in[9] = cluster_idx: partition of [0,N) -> mathematically unused.

    const int N = in_sizes[0] / HDIM;
    const int E = in_sizes[6];

    float* ws = (float*)d_ws;
    size_t off = 0;
    float* deg_inv = ws + off; off += (size_t)N;
    float* agg     = ws + off; off += (size_t)N * HDIM;
    float* featp   = ws + off; off += (size_t)N * HDIM;
    float* h1      = ws + off; off += (size_t)N * HDIM;
    float* posb    = ws + off; off += (size_t)N * HDIM;
    float* negb    = ws + off; off += (size_t)N * HDIM;
    float* colsum  = ws + off; off += 32;
    float* svec    = ws + off; off += 32;

    // degree + inverse
    hipMemsetAsync(deg_inv, 0, (size_t)N * sizeof(float), stream);
    deg_kernel<<<(E + 255) / 256, 256, 0, stream>>>(dst, deg_inv, E);
    inv_kernel<<<(N + 255) / 256, 256, 0, stream>>>(deg_inv, N);

    // corrupted input
    gather_perm_kernel<<<(N * 8 + 255) / 256, 256, 0, stream>>>(feature, perm,
                                                                featp, N);

    const float* h0s[2] = {feature, featp};
    float* outs[2]      = {posb, negb};
    const int aggGrid   = (E * 8 + 255) / 256;
    const int gemmGrid  = N / 64;              // 4 waves/block, 1 wave = 16 rows

    for (int p = 0; p < 2; ++p) {
        hipMemsetAsync(agg, 0, (size_t)N * HDIM * sizeof(float), stream);
        edge_agg_kernel<<<aggGrid, 256, 0, stream>>>(h0s[p], src, dst, agg, E);
        gin_layer_wmma<<<gemmGrid, 128, 0, stream>>>(h0s[p], agg, deg_inv,
                                                     W1, b1, h1);
        hipMemsetAsync(agg, 0, (size_t)N * HDIM * sizeof(float), stream);
        edge_agg_kernel<<<aggGrid, 256, 0, stream>>>(h1, src, dst, agg, E);
        gin_layer_wmma<<<gemmGrid, 128, 0, stream>>>(h1, agg, deg_inv,
                                                     W2, b2, outs[p]);
    }

    // summary / discriminator vector
    hipMemsetAsync(colsum, 0, 32 * sizeof(float), stream);
    colsum_kernel<<<512, 256, 0, stream>>>(posb, colsum, N);
    summary_s_kernel<<<1, 32, 0, stream>>>(colsum, Wd, svec, N);

    // final scalar loss
    hipMemsetAsync(d_out, 0, (size_t)out_size * sizeof(float), stream);
    loss_kernel<<<512, 256, 0, stream>>>(posb, negb, svec, (float*)d_out, N);
}